// BatchDynamicSoftLabelAssigner_70909910057788
// MI455X (gfx1250) — compile-verified
//
#include <hip/hip_runtime.h>
#include <float.h>
#include <math.h>

#define NCLS 80
#define TOPK 13
#define TPB_COL 256
#define TILE_P 128
#define LROW 113   // padded LDS row stride (gcd(113,64)=1 -> conflict-free)
#define GMAX 128

typedef float v2f __attribute__((ext_vector_type(2)));
typedef float v8f __attribute__((ext_vector_type(8)));

// ---------------- shared math helpers (must be identical across kernels) ----

__device__ __forceinline__ float iou_pair(float dx0, float dy0, float dx1, float dy1,
                                          float gx0, float gy0, float gx1, float gy1) {
  float a1 = fmaxf(dx1 - dx0, 0.f) * fmaxf(dy1 - dy0, 0.f);
  float a2 = fmaxf(gx1 - gx0, 0.f) * fmaxf(gy1 - gy0, 0.f);
  float ltx = fmaxf(dx0, gx0), lty = fmaxf(dy0, gy0);
  float rbx = fminf(dx1, gx1), rby = fminf(dy1, gy1);
  float inter = fmaxf(rbx - ltx, 0.f) * fmaxf(rby - lty, 0.f);
  float uni = fmaxf(a1 + a2 - inter, 1e-6f);
  return inter / uni;
}

__device__ __forceinline__ float cost_fn(float iou, float logit,
                                         float px, float py, float pst,
                                         float gx0, float gy0, float gx1, float gy1,
                                         int validp) {
  float gcx = (gx0 + gx1) * 0.5f, gcy = (gy0 + gy1) * 0.5f;
  float ddx = px - gcx, ddy = py - gcy;
  float dist = sqrtf(ddx * ddx + ddy * ddy) / pst * (validp ? 1.f : 0.f);
  float sc = exp2f((dist - 3.0f) * 3.321928094887362f);     // 10^(dist-3)
  float iouc = -logf(iou + 1e-7f) * 3.0f;                    // IOU_WEIGHT
  float sig = 1.0f / (1.0f + expf(-logit));
  float sp = fmaxf(logit, 0.f) + log1pf(expf(-fabsf(logit))); // stable softplus
  float bce = sp - logit * iou;
  float dd = iou - sig;
  float c = bce * dd * dd + iouc + sc;
  return validp ? c : 1e8f;                                  // INF
}

// ---------------- kernel 0: valid[b,p] = any g (strictly inside & pad) ------

__global__ void k_valid(const float* __restrict__ priors,
                        const float* __restrict__ gtb,
                        const unsigned char* __restrict__ pad,
                        int* __restrict__ valid, int B, int P, int G) {
  int idx = blockIdx.x * blockDim.x + threadIdx.x;
  if (idx >= B * P) return;
  int b = idx / P, p = idx - b * P;
  float px = priors[p * 4 + 0], py = priors[p * 4 + 1];
  int v = 0;
  const float* gb = gtb + (size_t)b * G * 4;
  const unsigned char* pd = pad + (size_t)b * G;
  for (int g = 0; g < G; ++g) {
    int inb = (px > gb[g * 4 + 0]) & (py > gb[g * 4 + 1]) &
              (gb[g * 4 + 2] > px) & (gb[g * 4 + 3] > py);
    v |= inb & (pd[g] ? 1 : 0);
  }
  valid[idx] = v;
}

// ---------------- kernel 1: dyn_ks[b,g] = clamp(int(sum top-13 IoU),1,..) ---

__global__ void k_dynks(const float* __restrict__ dec,
                        const float* __restrict__ gtb,
                        const unsigned char* __restrict__ pad,
                        int* __restrict__ dynks, int B, int P, int G) {
  int col = blockIdx.x;
  int b = col / G;
  int t = threadIdx.x;
  if (!pad[col]) { if (t == 0) dynks[col] = 0; return; }
  float gx0 = gtb[col * 4 + 0], gy0 = gtb[col * 4 + 1];
  float gx1 = gtb[col * 4 + 2], gy1 = gtb[col * 4 + 3];
  float tk[TOPK];
#pragma unroll
  for (int j = 0; j < TOPK; ++j) tk[j] = -1.f;   // descending, tk[0]=max
  const float* db = dec + (size_t)b * P * 4;
  for (int p = t; p < P; p += TPB_COL) {
    float v = iou_pair(db[p * 4 + 0], db[p * 4 + 1], db[p * 4 + 2], db[p * 4 + 3],
                       gx0, gy0, gx1, gy1);
#pragma unroll
    for (int j = 0; j < TOPK; ++j) {
      if (v > tk[j]) { float t0 = tk[j]; tk[j] = v; v = t0; }
    }
  }
  __shared__ float sm[TPB_COL * TOPK];
#pragma unroll
  for (int j = 0; j < TOPK; ++j) sm[t * TOPK + j] = tk[j];
  for (int s = TPB_COL / 2; s > 0; s >>= 1) {
    __syncthreads();
    if (t < s) {
#pragma unroll
      for (int j = 0; j < TOPK; ++j) {
        float v = sm[(t + s) * TOPK + j];
#pragma unroll
        for (int q = 0; q < TOPK; ++q) {
          if (v > tk[q]) { float t0 = tk[q]; tk[q] = v; v = t0; }
        }
      }
#pragma unroll
      for (int j = 0; j < TOPK; ++j) sm[t * TOPK + j] = tk[j];
    }
  }
  if (t == 0) {
    float s = 0.f;
#pragma unroll
    for (int j = 0; j < TOPK; ++j) s += tk[j];
    int k = (int)s;                 // trunc toward zero, sum >= 0
    if (k < 1) k = 1;
    dynks[col] = k;
  }
}

// ---- kernel 2: per column, dyn_ks-th smallest (cost,idx) lexicographic ----
// Reproduces stable double-argsort ranks: rank[p] < k  <=>  (cost,p) <=lex kth pair.

__global__ void k_thresh(const float* __restrict__ pred,
                         const float* __restrict__ priors,
                         const float* __restrict__ dec,
                         const float* __restrict__ gtb,
                         const int* __restrict__ glabels,
                         const unsigned char* __restrict__ pad,
                         const int* __restrict__ valid,
                         const int* __restrict__ dynks,
                         float* __restrict__ tval, int* __restrict__ tidx,
                         int B, int P, int G) {
  int col = blockIdx.x;
  int b = col / G;
  int t = threadIdx.x;
  if (!pad[col]) { if (t == 0) { tval[col] = -1e30f; tidx[col] = -1; } return; }
  float gx0 = gtb[col * 4 + 0], gy0 = gtb[col * 4 + 1];
  float gx1 = gtb[col * 4 + 2], gy1 = gtb[col * 4 + 3];
  int lab = glabels[col];
  float tv[TOPK]; int ti[TOPK];
#pragma unroll
  for (int j = 0; j < TOPK; ++j) { tv[j] = FLT_MAX; ti[j] = 0x7FFFFFFF; }
  const float* db = dec + (size_t)b * P * 4;
  for (int p = t; p < P; p += TPB_COL) {
    float iou = iou_pair(db[p * 4 + 0], db[p * 4 + 1], db[p * 4 + 2], db[p * 4 + 3],
                         gx0, gy0, gx1, gy1);
    float logit = pred[((size_t)b * P + p) * NCLS + lab];
    float v = cost_fn(iou, logit, priors[p * 4 + 0], priors[p * 4 + 1], priors[p * 4 + 2],
                      gx0, gy0, gx1, gy1, valid[b * P + p]);
    int vi = p;
#pragma unroll
    for (int j = 0; j < TOPK; ++j) {
      bool lt = (v < tv[j]) || (v == tv[j] && vi < ti[j]);
      if (lt) { float t0 = tv[j]; int i0 = ti[j]; tv[j] = v; ti[j] = vi; v = t0; vi = i0; }
    }
  }
  __shared__ float smv[TPB_COL * TOPK];
  __shared__ int   smi[TPB_COL * TOPK];
#pragma unroll
  for (int j = 0; j < TOPK; ++j) { smv[t * TOPK + j] = tv[j]; smi[t * TOPK + j] = ti[j]; }
  for (int s = TPB_COL / 2; s > 0; s >>= 1) {
    __syncthreads();
    if (t < s) {
#pragma unroll
      for (int j = 0; j < TOPK; ++j) {
        float v = smv[(t + s) * TOPK + j];
        int vi = smi[(t + s) * TOPK + j];
#pragma unroll
        for (int q = 0; q < TOPK; ++q) {
          bool lt = (v < tv[q]) || (v == tv[q] && vi < ti[q]);
          if (lt) { float t0 = tv[q]; int i0 = ti[q]; tv[q] = v; ti[q] = vi; v = t0; vi = i0; }
        }
      }
#pragma unroll
      for (int j = 0; j < TOPK; ++j) { smv[t * TOPK + j] = tv[j]; smi[t * TOPK + j] = ti[j]; }
    }
  }
  if (t == 0) {
    int k = dynks[col];              // 1..13
    float Tv = tv[TOPK - 1]; int Ti = ti[TOPK - 1];
#pragma unroll
    for (int j = 0; j < TOPK; ++j) { if (j == k - 1) { Tv = tv[j]; Ti = ti[j]; } }
    tval[col] = Tv; tidx[col] = Ti;
  }
}

// ---- kernel 3: WMMA one-hot logits gather + per-prior assignment ----------

__global__ void __launch_bounds__(TILE_P)
k_assign(const float* __restrict__ pred, const float* __restrict__ priors,
         const float* __restrict__ dec, const float* __restrict__ gtb,
         const int* __restrict__ glabels, const int* __restrict__ valid,
         const float* __restrict__ tval, const int* __restrict__ tidx,
         float* __restrict__ out, int B, int P, int G) {
  __shared__ float s_log[TILE_P * LROW];
  __shared__ float s_gtb[GMAX * 4];
  __shared__ int   s_lab[GMAX];
  __shared__ float s_tv[GMAX];
  __shared__ int   s_ti[GMAX];
  __shared__ int   s_plab[TILE_P];
  __shared__ float s_miou[TILE_P];
  __shared__ int   s_mg[TILE_P];
  __shared__ int   s_fg[TILE_P];

  int nPt = (P + TILE_P - 1) / TILE_P;
  int b = blockIdx.x / nPt;
  int p0 = (blockIdx.x - b * nPt) * TILE_P;
  int t = threadIdx.x;
  int Gc = G < GMAX ? G : GMAX;

  for (int i = t; i < Gc; i += TILE_P) {
    s_lab[i] = glabels[b * G + i];
    s_tv[i] = tval[b * G + i];
    s_ti[i] = tidx[b * G + i];
    s_gtb[i * 4 + 0] = gtb[((size_t)b * G + i) * 4 + 0];
    s_gtb[i * 4 + 1] = gtb[((size_t)b * G + i) * 4 + 1];
    s_gtb[i * 4 + 2] = gtb[((size_t)b * G + i) * 4 + 2];
    s_gtb[i * 4 + 3] = gtb[((size_t)b * G + i) * 4 + 3];
  }
  __syncthreads();

  // Phase 1: logits[p][g] = pred[b,p,label[g]]  as one-hot GEMM (exact in f32)
#if defined(__gfx1250__) && __has_builtin(__builtin_amdgcn_wmma_f32_16x16x4_f32)
  {
    int wave = t >> 5, lane = t & 31;
    int half = (lane < 16) ? 0 : 1;        // A/B: lanes 16-31 hold K=2,3
    int mrow = lane & 15;
    int koff = half * 2;
    int nGt = (Gc + 15) >> 4;
    for (int mt = wave; mt < TILE_P / 16; mt += TILE_P / 32) {
      int prow = p0 + mt * 16 + mrow;
      int prc = prow < P ? prow : P - 1;   // clamp; tail rows discarded later
      const float* arow = pred + ((size_t)b * P + prc) * NCLS;
      v2f af[NCLS / 4];
#pragma unroll
      for (int k = 0; k < NCLS / 4; ++k) {
        af[k].x = arow[k * 4 + koff];
        af[k].y = arow[k * 4 + koff + 1];
      }
      for (int nt = 0; nt < nGt; ++nt) {
        int gcol = nt * 16 + mrow;
        int lab = (gcol < Gc) ? s_lab[gcol] : -1;
        v8f acc = {};
#pragma unroll
        for (int k = 0; k < NCLS / 4; ++k) {
          int kb = k * 4 + koff;
          v2f bf;
          bf.x = (lab == kb) ? 1.f : 0.f;
          bf.y = (lab == kb + 1) ? 1.f : 0.f;
          acc = __builtin_amdgcn_wmma_f32_16x16x4_f32(
              false, af[k], false, bf, (short)0, acc, false, false);
        }
#pragma unroll
        for (int r = 0; r < 8; ++r) {
          int M = r + half * 8;            // C/D layout: VGPR r -> rows r, r+8
          s_log[(mt * 16 + M) * LROW + nt * 16 + mrow] = acc[r];
        }
      }
    }
  }
#else
  for (int e = t; e < TILE_P * Gc; e += TILE_P) {
    int pl = e / Gc, g = e - pl * Gc;
    int prow = p0 + pl;
    int prc = prow < P ? prow : P - 1;
    s_log[pl * LROW + g] = pred[((size_t)b * P + prc) * NCLS + s_lab[g]];
  }
#endif
  __syncthreads();

  // Phase 2: per-prior matching / resolution
  int p = p0 + t;
  int lab_out = NCLS, mg = 0, fgv = 0;
  float miou = 0.f;
  if (p < P) {
    int vld = valid[b * P + p];
    const float* dbp = dec + ((size_t)b * P + p) * 4;
    float dx0 = dbp[0], dy0 = dbp[1], dx1 = dbp[2], dy1 = dbp[3];
    float px = priors[p * 4 + 0], py = priors[p * 4 + 1], pst = priors[p * 4 + 2];
    int cnt = 0, firstG = 0, bestG = 0;
    float firstIou = 0.f, bestC = 0.f, bestIou = 0.f;
    for (int g = 0; g < Gc; ++g) {
      float gx0 = s_gtb[g * 4 + 0], gy0 = s_gtb[g * 4 + 1];
      float gx1 = s_gtb[g * 4 + 2], gy1 = s_gtb[g * 4 + 3];
      float iou = iou_pair(dx0, dy0, dx1, dy1, gx0, gy0, gx1, gy1);
      float logit = s_log[t * LROW + g];
      float c = cost_fn(iou, logit, px, py, pst, gx0, gy0, gx1, gy1, vld);
      if (g == 0 || c < bestC) { bestC = c; bestG = g; bestIou = iou; } // argmin, first index
      bool m = (c < s_tv[g]) || (c == s_tv[g] && p <= s_ti[g]);        // rank < dyn_ks & pad
      if (m) { if (cnt == 0) { firstG = g; firstIou = iou; } ++cnt; }
    }
    if (cnt > 1)       { mg = bestG;  miou = bestIou;  fgv = 1; }
    else if (cnt == 1) { mg = firstG; miou = firstIou; fgv = 1; }
    if (fgv) lab_out = s_lab[mg];
  }
  s_plab[t] = lab_out; s_miou[t] = miou; s_mg[t] = mg; s_fg[t] = fgv;
  __syncthreads();

  // Phase 3: coalesced output writes. d_out = [labels | boxes | scores | fg]
  size_t BP = (size_t)B * P;
  float* outL = out;
  float* outB = out + BP;
  float* outS = out + BP * 5;
  float* outF = out + BP * (5 + NCLS);
  if (p < P) {
    outL[(size_t)b * P + p] = (float)s_plab[t];
    outF[(size_t)b * P + p] = (float)s_fg[t];
  }
  for (int e = t; e < TILE_P * 4; e += TILE_P) {
    int pl = e >> 2, c = e & 3;
    int pp = p0 + pl;
    if (pp < P) {
      float v = s_fg[pl] ? s_gtb[s_mg[pl] * 4 + c] : 0.f;
      outB[((size_t)b * P + pp) * 4 + c] = v;
    }
  }
  for (int e = t; e < TILE_P * NCLS; e += TILE_P) {
    int pl = e / NCLS, c = e - pl * NCLS;
    int pp = p0 + pl;
    if (pp < P) {
      float v = (s_fg[pl] && c == s_plab[pl]) ? s_miou[pl] : 0.f;
      outS[((size_t)b * P + pp) * NCLS + c] = v;
    }
  }
}

// ---------------------------------------------------------------------------

extern "C" void kernel_launch(void* const* d_in, const int* in_sizes, int n_in,
                              void* d_out, int out_size, void* d_ws, size_t ws_size,
                              hipStream_t stream) {
  (void)n_in; (void)out_size; (void)ws_size;
  int P = in_sizes[1] / 4;
  int B = in_sizes[2] / (4 * P);
  int G = in_sizes[4] / B;

  const float* pred   = (const float*)d_in[0];
  const float* priors = (const float*)d_in[1];
  const float* dec    = (const float*)d_in[2];
  const float* gtb    = (const float*)d_in[3];
  const int*   glab   = (const int*)d_in[4];
  const unsigned char* pad = (const unsigned char*)d_in[5];

  int*   d_valid = (int*)d_ws;                       // B*P
  int*   d_dynks = d_valid + (size_t)B * P;          // B*G
  float* d_tval  = (float*)(d_dynks + (size_t)B * G);// B*G
  int*   d_tidx  = (int*)(d_tval + (size_t)B * G);   // B*G

  k_valid<<<(B * P + 255) / 256, 256, 0, stream>>>(priors, gtb, pad, d_valid, B, P, G);
  k_dynks<<<B * G, TPB_COL, 0, stream>>>(dec, gtb, pad, d_dynks, B, P, G);
  k_thresh<<<B * G, TPB_COL, 0, stream>>>(pred, priors, dec, gtb, glab, pad,
                                          d_valid, d_dynks, d_tval, d_tidx, B, P, G);
  int nPt = (P + TILE_P - 1) / TILE_P;
  k_assign<<<B * nPt, TILE_P, 0, stream>>>(pred, priors, dec, gtb, glab,
                                           d_valid, d_tval, d_tidx,
                                           (float*)d_out, B, P, G);
}